// MoELayer_79637283602980
// MI455X (gfx1250) — compile-verified
//
#include <hip/hip_runtime.h>
#include <stdint.h>

#define T_TOK 8192
#define DIM   1024
#define NE    8
#define HID   2048
#define OUTD  1024

#define BM 64
#define BN 128
#define KB 16
#define ASTRIDE 20    // KB + 4 pad (16B-aligned rows, bank-spread)
#define BSTRIDE 136   // BN + 8 pad (16B-aligned rows, bank-spread)

typedef float v2f __attribute__((ext_vector_type(2)));
typedef float v8f __attribute__((ext_vector_type(8)));

__device__ __forceinline__ v8f wmma_f32(v2f a, v2f b, v8f c) {
    // 8 args: (neg_a, A, neg_b, B, c_mod, C, reuse_a, reuse_b)
    return __builtin_amdgcn_wmma_f32_16x16x4_f32(false, a, false, b, (short)0, c,
                                                 false, false);
}

// Async global -> LDS copy of 16 bytes per lane (CDNA5 path, ASYNCcnt-tracked).
__device__ __forceinline__ void async_copy_b128(uint32_t lds_off, const float* gptr) {
    asm volatile("global_load_async_to_lds_b128 %0, %1, off"
                 :: "v"(lds_off), "v"((uint64_t)(uintptr_t)gptr)
                 : "memory");
}

__device__ __forceinline__ void wait_async0() {
    asm volatile("s_wait_asynccnt 0x0" ::: "memory");
}

__device__ __forceinline__ uint32_t lds_offset(const void* p) {
    // generic pointer to LDS keeps the LDS byte offset in the low 32 bits
    return (uint32_t)(uintptr_t)p;
}

// ---------------------------------------------------------------------------
// Kernel 1: fp32 gating + argmax + per-expert token list build (atomic append)
// One wave32 per token; exact fp32 so routing matches the fp32 reference.
// ---------------------------------------------------------------------------
__global__ __launch_bounds__(256) void gate_kernel(
    const float* __restrict__ x, const float* __restrict__ gw,
    const float* __restrict__ gb, int* __restrict__ counts,
    int* __restrict__ lists) {
    const int lane  = threadIdx.x & 31;
    const int token = blockIdx.x * 8 + (threadIdx.x >> 5);

    float acc[NE];
#pragma unroll
    for (int e = 0; e < NE; ++e) acc[e] = 0.f;

    const float* xrow = x + (size_t)token * DIM;
    for (int d = lane; d < DIM; d += 32) {
        float xv = xrow[d];
        float4 g0 = *(const float4*)(gw + d * NE);
        float4 g1 = *(const float4*)(gw + d * NE + 4);
        acc[0] += xv * g0.x; acc[1] += xv * g0.y;
        acc[2] += xv * g0.z; acc[3] += xv * g0.w;
        acc[4] += xv * g1.x; acc[5] += xv * g1.y;
        acc[6] += xv * g1.z; acc[7] += xv * g1.w;
    }
#pragma unroll
    for (int e = 0; e < NE; ++e) {
#pragma unroll
        for (int off = 16; off > 0; off >>= 1)
            acc[e] += __shfl_xor(acc[e], off, 32);
    }
    if (lane == 0) {
        int best = 0;
        float bv = acc[0] + gb[0];
#pragma unroll
        for (int e = 1; e < NE; ++e) {
            float v = acc[e] + gb[e];
            if (v > bv) { bv = v; best = e; }   // strict > keeps first max (jnp.argmax)
        }
        int pos = atomicAdd(&counts[best], 1);
        lists[best * T_TOK + pos] = token;
    }
}

// ---------------------------------------------------------------------------
// Kernel 2: grouped expert GEMM (gather by list), y = x_g @ W_e + b_e
// 64x128 block tile, double-buffered LDS fed by async global->LDS copies,
// wave32 computes 32x32 via four 16x16 fp32 WMMA accumulators.
// ---------------------------------------------------------------------------
__global__ __launch_bounds__(256) void expert_gemm_kernel(
    const float* __restrict__ x, const float* __restrict__ ew,
    const float* __restrict__ eb, const int* __restrict__ counts,
    const int* __restrict__ lists, float* __restrict__ y) {
    __shared__ float As[2][BM * ASTRIDE];
    __shared__ float Bs[2][KB * BSTRIDE];
    __shared__ int   toks[BM];

    const int e     = blockIdx.z;
    const int count = counts[e];
    const int row0  = blockIdx.y * BM;
    if (row0 >= count) return;                 // block-uniform: EXEC stays full
    const int n0 = blockIdx.x * BN;

    const int tid = threadIdx.x;
    if (tid < BM) {
        int r = row0 + tid;
        toks[tid] = (r < count) ? lists[e * T_TOK + r] : -1;
    }
    __syncthreads();

    const int wid  = tid >> 5;
    const int lane = tid & 31;
    const int wm   = wid >> 2;       // 0..1 : 32-row slab
    const int wn   = wid & 3;        // 0..3 : 32-col slab
    const int half = lane >> 4;      // ISA fragment half-wave split
    const int ml   = lane & 15;

    // Stage-copy source mapping (one b128 for A, two b128 for B per thread).
    const int ar = tid >> 2;          // 0..63
    const int ac = (tid & 3) * 4;
    const int bk = tid >> 4;          // 0..15
    const int bn = (tid & 15) * 8;

    // Padded rows: any valid address works -- row m only affects output row m,
    // and padded rows are never stored. Clamp to a real token for the copy.
    int tokA = toks[ar];
    if (tokA < 0) tokA = toks[0];    // toks[0] valid because row0 < count
    const float* aSrc = x + (size_t)tokA * DIM + ac;
    const float* bSrc = ew + ((size_t)e * DIM + bk) * HID + n0 + bn;

    v8f acc00 = {}, acc01 = {}, acc10 = {}, acc11 = {};

    // prologue: stage k0 = 0 into buffer 0
    {
        async_copy_b128(lds_offset(&As[0][ar * ASTRIDE + ac]), aSrc);
        async_copy_b128(lds_offset(&Bs[0][bk * BSTRIDE + bn]), bSrc);
        async_copy_b128(lds_offset(&Bs[0][bk * BSTRIDE + bn + 4]), bSrc + 4);
    }
    wait_async0();
    __syncthreads();

    for (int k0 = 0; k0 < DIM; k0 += KB) {
        const int buf = (k0 / KB) & 1;
        if (k0 + KB < DIM) {   // stage next K-slab into the other buffer
            const int nb = buf ^ 1;
            async_copy_b128(lds_offset(&As[nb][ar * ASTRIDE + ac]), aSrc + (k0 + KB));
            const float* bp = bSrc + (size_t)(k0 + KB) * HID;
            async_copy_b128(lds_offset(&Bs[nb][bk * BSTRIDE + bn]), bp);
            async_copy_b128(lds_offset(&Bs[nb][bk * BSTRIDE + bn + 4]), bp + 4);
        }

        const float* Ab = As[buf];
        const float* Bb = Bs[buf];
#pragma unroll
        for (int kk = 0; kk < KB; kk += 4) {
            v2f a0 = *(const v2f*)(&Ab[(wm * 32 + ml)      * ASTRIDE + kk + 2 * half]);
            v2f a1 = *(const v2f*)(&Ab[(wm * 32 + 16 + ml) * ASTRIDE + kk + 2 * half]);
            int c0 = wn * 32 + ml;
            v2f b0, b1;
            b0.x = Bb[(kk + 2 * half)     * BSTRIDE + c0];
            b0.y = Bb[(kk + 2 * half + 1) * BSTRIDE + c0];
            b1.x = Bb[(kk + 2 * half)     * BSTRIDE + c0 + 16];
            b1.y = Bb[(kk + 2 * half + 1) * BSTRIDE + c0 + 16];
            acc00 = wmma_f32(a0, b0, acc00);
            acc01 = wmma_f32(a0, b1, acc01);
            acc10 = wmma_f32(a1, b0, acc10);
            acc11 = wmma_f32(a1, b1, acc11);
        }
        wait_async0();
        __syncthreads();
    }

    const int col0  = n0 + wn * 32 + ml;
    const float bias0 = eb[e * HID + col0];
    const float bias1 = eb[e * HID + col0 + 16];
#pragma unroll
    for (int r = 0; r < 8; ++r) {
        int m0 = wm * 32 + r + 8 * half;        // sub_m = 0
        int m1 = wm * 32 + 16 + r + 8 * half;   // sub_m = 1
        int t0 = toks[m0], t1 = toks[m1];
        if (t0 >= 0) {
            y[(size_t)t0 * HID + col0]      = acc00[r] + bias0;
            y[(size_t)t0 * HID + col0 + 16] = acc01[r] + bias1;
        }
        if (t1 >= 0) {
            y[(size_t)t1 * HID + col0]      = acc10[r] + bias0;
            y[(size_t)t1 * HID + col0 + 16] = acc11[r] + bias1;
        }
    }
}

// ---------------------------------------------------------------------------
// Kernel 3: out = y @ out_w + out_b   (dense 8192x2048x1024, same pipeline)
// ---------------------------------------------------------------------------
__global__ __launch_bounds__(256) void out_gemm_kernel(
    const float* __restrict__ y, const float* __restrict__ ow,
    const float* __restrict__ ob, float* __restrict__ out) {
    __shared__ float As[2][BM * ASTRIDE];
    __shared__ float Bs[2][KB * BSTRIDE];

    const int row0 = blockIdx.y * BM;
    const int n0   = blockIdx.x * BN;
    const int tid  = threadIdx.x;
    const int wid  = tid >> 5;
    const int lane = tid & 31;
    const int wm   = wid >> 2;
    const int wn   = wid & 3;
    const int half = lane >> 4;
    const int ml   = lane & 15;

    const int ar = tid >> 2;
    const int ac = (tid & 3) * 4;
    const int bk = tid >> 4;
    const int bn = (tid & 15) * 8;

    const float* aSrc = y + (size_t)(row0 + ar) * HID + ac;
    const float* bSrc = ow + (size_t)bk * OUTD + n0 + bn;

    v8f acc00 = {}, acc01 = {}, acc10 = {}, acc11 = {};

    {
        async_copy_b128(lds_offset(&As[0][ar * ASTRIDE + ac]), aSrc);
        async_copy_b128(lds_offset(&Bs[0][bk * BSTRIDE + bn]), bSrc);
        async_copy_b128(lds_offset(&Bs[0][bk * BSTRIDE + bn + 4]), bSrc + 4);
    }
    wait_async0();
    __syncthreads();

    for (int k0 = 0; k0 < HID; k0 += KB) {
        const int buf = (k0 / KB) & 1;
        if (k0 + KB < HID) {
            const int nb = buf ^ 1;
            async_copy_b128(lds_offset(&As[nb][ar * ASTRIDE + ac]), aSrc + (k0 + KB));
            const float* bp = bSrc + (size_t)(k0 + KB) * OUTD;
            async_copy_b128(lds_offset(&Bs[nb][bk * BSTRIDE + bn]), bp);
            async_copy_b128(lds_offset(&Bs[nb][bk * BSTRIDE + bn + 4]), bp + 4);
        }

        const float* Ab = As[buf];
        const float* Bb = Bs[buf];
#pragma unroll
        for (int kk = 0; kk < KB; kk += 4) {
            v2f a0 = *(const v2f*)(&Ab[(wm * 32 + ml)      * ASTRIDE + kk + 2 * half]);
            v2f a1 = *(const v2f*)(&Ab[(wm * 32 + 16 + ml) * ASTRIDE + kk + 2 * half]);
            int c0 = wn * 32 + ml;
            v2f b0, b1;
            b0.x = Bb[(kk + 2 * half)     * BSTRIDE + c0];
            b0.y = Bb[(kk + 2 * half + 1) * BSTRIDE + c0];
            b1.x = Bb[(kk + 2 * half)     * BSTRIDE + c0 + 16];
            b1.y = Bb[(kk + 2 * half + 1) * BSTRIDE + c0 + 16];
            acc00 = wmma_f32(a0, b0, acc00);
            acc01 = wmma_f32(a0, b1, acc01);
            acc10 = wmma_f32(a1, b0, acc10);
            acc11 = wmma_f32(a1, b1, acc11);
        }
        wait_async0();
        __syncthreads();
    }

    const int col0  = n0 + wn * 32 + ml;
    const float bias0 = ob[col0];
    const float bias1 = ob[col0 + 16];
#pragma unroll
    for (int r = 0; r < 8; ++r) {
        int m0 = row0 + wm * 32 + r + 8 * half;
        int m1 = m0 + 16;
        out[(size_t)m0 * OUTD + col0]      = acc00[r] + bias0;
        out[(size_t)m0 * OUTD + col0 + 16] = acc01[r] + bias1;
        out[(size_t)m1 * OUTD + col0]      = acc10[r] + bias0;
        out[(size_t)m1 * OUTD + col0 + 16] = acc11[r] + bias1;
    }
}

extern "C" void kernel_launch(void* const* d_in, const int* in_sizes, int n_in,
                              void* d_out, int out_size, void* d_ws, size_t ws_size,
                              hipStream_t stream) {
    (void)in_sizes; (void)n_in; (void)out_size; (void)ws_size;
    const float* x  = (const float*)d_in[0];
    const float* gw = (const float*)d_in[1];
    const float* gb = (const float*)d_in[2];
    const float* ew = (const float*)d_in[3];
    const float* eb = (const float*)d_in[4];
    const float* ow = (const float*)d_in[5];
    const float* ob = (const float*)d_in[6];
    float* out = (float*)d_out;

    char* ws     = (char*)d_ws;
    int*  counts = (int*)ws;                    // 8 ints
    int*  lists  = (int*)(ws + 256);            // E*T ints = 256 KB
    float* y     = (float*)(ws + (1 << 20));    // T*H floats = 64 MB

    hipMemsetAsync(counts, 0, NE * sizeof(int), stream);
    gate_kernel<<<T_TOK / 8, 256, 0, stream>>>(x, gw, gb, counts, lists);
    expert_gemm_kernel<<<dim3(HID / BN, T_TOK / BM, NE), 256, 0, stream>>>(
        x, ew, eb, counts, lists, y);
    out_gemm_kernel<<<dim3(OUTD / BN, T_TOK / BM), 256, 0, stream>>>(y, ow, ob, out);
}